// AutoLink_l3scale_33998961116073
// MI455X (gfx1250) — compile-verified
//
#include <hip/hip_runtime.h>

#define N_NODES 50000
#define N_EDGES 800000
#define DIM 256
#define OUT_STRIDE (3 * DIM)   // [N, 3, D]
#define BM 32                  // rows per block tile
#define KTOT 512               // concat K = 2*DIM

typedef __attribute__((ext_vector_type(2))) float v2f;
typedef __attribute__((ext_vector_type(8))) float v8f;

// ---------------------------------------------------------------- utilities
__global__ __launch_bounds__(256) void zero4_kernel(float4* __restrict__ p, int n4) {
    int i = blockIdx.x * blockDim.x + threadIdx.x;
    if (i < n4) p[i] = make_float4(0.f, 0.f, 0.f, 0.f);
}

__global__ __launch_bounds__(256) void count_kernel(const int* __restrict__ dst,
                                                    float* __restrict__ cnt) {
    int e = blockIdx.x * blockDim.x + threadIdx.x;
    if (e < N_EDGES) atomicAdd(&cnt[dst[e]], 1.0f);
}

__global__ __launch_bounds__(256) void invcnt_kernel(float* __restrict__ cnt) {
    int i = blockIdx.x * blockDim.x + threadIdx.x;
    if (i < N_NODES) cnt[i] = 1.0f / fmaxf(cnt[i], 1.0f);
}

// ------------------------------------------------------- edge scatter (mean numerator)
// 64 threads per edge; each thread handles 4 consecutive dims:
// one b128 gather + 4 coalesced global f32 atomics.
__global__ __launch_bounds__(256) void scatter_kernel(const float* __restrict__ h,
                                                      const int* __restrict__ src,
                                                      const int* __restrict__ dst,
                                                      float* __restrict__ agg) {
    int t = blockIdx.x * blockDim.x + threadIdx.x;
    int e = t >> 6;                // wave-uniform (32 lanes cover half an edge's dims)
    int c = (t & 63) << 2;         // dim offset, 4 floats per thread
    if (e >= N_EDGES) return;
    int s = __builtin_amdgcn_readfirstlane(src[e]);
    int d = __builtin_amdgcn_readfirstlane(dst[e]);
    const float4 v = *(const float4*)(h + (size_t)s * DIM + c);
    float* a = agg + (size_t)d * DIM + c;
    atomicAdd(a + 0, v.x);
    atomicAdd(a + 1, v.y);
    atomicAdd(a + 2, v.z);
    atomicAdd(a + 3, v.w);
}

// ------------------------------------------------------- fused SAGE layer GEMM
// out_row = (agg/max(cnt,1)) @ Wl + h @ Wr + b  ==  [mean, h] (K=512) @ [Wl; Wr]
// Block: 256 thr = 8 waves, tile 32 rows x 256 cols.
// Wave w: cols [32w, 32w+32), 4 accumulators (2 row-tiles x 2 col-tiles).
// A staged in 64KB LDS, XOR-swizzled for conflict-free ds_load_b64 fragments.
__global__ __launch_bounds__(256) void sage_gemm_kernel(const float* __restrict__ agg,
                                                        const float* __restrict__ invc,
                                                        const float* __restrict__ hin,
                                                        const float* __restrict__ Wl,
                                                        const float* __restrict__ Wr,
                                                        const float* __restrict__ bias,
                                                        float* __restrict__ out,    // d_out + li*DIM
                                                        float* __restrict__ hnext)  // relu'd, or null
{
    __shared__ float At[BM * KTOT];   // 32*512*4 = 65536 bytes exactly
    const int tid = threadIdx.x;
    const int mbase = blockIdx.x * BM;

    // ---- stage A: 32 rows x 512 floats = 4096 float4 / 256 threads = 16 each
    #pragma unroll
    for (int it = 0; it < 16; ++it) {
        int f = it * 256 + tid;       // float4 index
        int row = f >> 7;             // 0..31
        int q = f & 127;              // float4 index within row
        int node = mbase + row;
        if (node >= N_NODES) node = N_NODES - 1;   // clamp; masked at store
        int k = q << 2;
        float4 v;
        if (k < DIM) {
            v = *(const float4*)(agg + (size_t)node * DIM + k);
            float ic = invc[node];
            v.x *= ic; v.y *= ic; v.z *= ic; v.w *= ic;
        } else {
            v = *(const float4*)(hin + (size_t)node * DIM + (k - DIM));
        }
        int sw = (q ^ (row & 15)) << 2;            // XOR-swizzled float offset
        *(float4*)(&At[row * KTOT + sw]) = v;
    }
    __syncthreads();

    const int lane = tid & 31;
    const int wv = tid >> 5;        // 0..7
    const int hi = lane >> 4;       // half-wave: K offset +2 (A), K-row +2 (B)
    const int ln = lane & 15;
    const int col0 = wv * 32 + ln;
    const int col1 = col0 + 16;

    v8f c00 = {}, c01 = {}, c10 = {}, c11 = {};

    // A fragment bases: row-tile0 row=ln, row-tile1 row=16+ln (same xor mask ln)
    const float* A0 = &At[ln * KTOT + 2 * hi];
    const float* A1 = &At[(16 + ln) * KTOT + 2 * hi];
    const int xr = ln;

    const float* Ws[2] = { Wl, Wr };
    #pragma unroll
    for (int half = 0; half < 2; ++half) {
        const float* __restrict__ W = Ws[half];
        #pragma unroll 4
        for (int kk = 0; kk < 64; ++kk) {
            const int kkg = half * 64 + kk;             // global K-step (K = 4*kkg)
            const int off = (kkg ^ xr) << 2;            // swizzled float offset
            v2f a0 = *(const v2f*)(A0 + off);
            v2f a1 = *(const v2f*)(A1 + off);
            // B fragment: lane ln -> col, half-wave -> K rows {0,1} / {2,3}
            const float* wb = W + (size_t)(kk * 4 + 2 * hi) * DIM;
            v2f b0; b0.x = wb[col0]; b0.y = wb[DIM + col0];
            v2f b1; b1.x = wb[col1]; b1.y = wb[DIM + col1];
            c00 = __builtin_amdgcn_wmma_f32_16x16x4_f32(false, a0, false, b0, (short)0, c00, false, false);
            c01 = __builtin_amdgcn_wmma_f32_16x16x4_f32(false, a0, false, b1, (short)0, c01, false, false);
            c10 = __builtin_amdgcn_wmma_f32_16x16x4_f32(false, a1, false, b0, (short)0, c10, false, false);
            c11 = __builtin_amdgcn_wmma_f32_16x16x4_f32(false, a1, false, b1, (short)0, c11, false, false);
        }
    }

    // ---- epilogue: bias, write pre-ReLU to out slice, ReLU'd to hnext
    const float bv0 = bias[col0];
    const float bv1 = bias[col1];
    #pragma unroll
    for (int r = 0; r < 8; ++r) {
        const int m = r + 8 * hi;                 // C layout: VGPR r -> M = r (+8 upper half)
        const int n0 = mbase + m;                 // row-tile 0
        const int n1 = mbase + 16 + m;            // row-tile 1
        if (n0 < N_NODES) {
            float v = c00[r] + bv0;
            out[(size_t)n0 * OUT_STRIDE + col0] = v;
            if (hnext) hnext[(size_t)n0 * DIM + col0] = fmaxf(v, 0.f);
            v = c01[r] + bv1;
            out[(size_t)n0 * OUT_STRIDE + col1] = v;
            if (hnext) hnext[(size_t)n0 * DIM + col1] = fmaxf(v, 0.f);
        }
        if (n1 < N_NODES) {
            float v = c10[r] + bv0;
            out[(size_t)n1 * OUT_STRIDE + col0] = v;
            if (hnext) hnext[(size_t)n1 * DIM + col0] = fmaxf(v, 0.f);
            v = c11[r] + bv1;
            out[(size_t)n1 * OUT_STRIDE + col1] = v;
            if (hnext) hnext[(size_t)n1 * DIM + col1] = fmaxf(v, 0.f);
        }
    }
}

// ---------------------------------------------------------------- launcher
extern "C" void kernel_launch(void* const* d_in, const int* in_sizes, int n_in,
                              void* d_out, int out_size, void* d_ws, size_t ws_size,
                              hipStream_t stream) {
    const float* x    = (const float*)d_in[0];
    const int*   esrc = (const int*)d_in[1];
    const int*   edst = (const int*)d_in[2];
    const float* Wl[3] = { (const float*)d_in[3], (const float*)d_in[6], (const float*)d_in[9]  };
    const float* Wr[3] = { (const float*)d_in[4], (const float*)d_in[7], (const float*)d_in[10] };
    const float* bb[3] = { (const float*)d_in[5], (const float*)d_in[8], (const float*)d_in[11] };
    float* out = (float*)d_out;

    // workspace layout (floats): cnt/inv | agg | hA | hB
    float* ws  = (float*)d_ws;
    float* cnt = ws;                                    // N_NODES (padded to 51200)
    float* agg = ws + 51200;                            // N*DIM
    float* hA  = agg + (size_t)N_NODES * DIM;           // N*DIM
    float* hB  = hA  + (size_t)N_NODES * DIM;           // N*DIM

    const int feat4   = N_NODES * DIM / 4;              // 3,200,000 float4
    const int cnt4    = (N_NODES + 3) / 4;

    // degree counts -> reciprocal (layer-invariant)
    zero4_kernel<<<(cnt4 + 255) / 256, 256, 0, stream>>>((float4*)cnt, cnt4);
    count_kernel<<<(N_EDGES + 255) / 256, 256, 0, stream>>>(edst, cnt);
    invcnt_kernel<<<(N_NODES + 255) / 256, 256, 0, stream>>>(cnt);

    const int scatterGrid = (N_EDGES * 64) / 256;       // 200000
    const int gemmGrid    = (N_NODES + BM - 1) / BM;    // 1563
    const int zeroGrid    = (feat4 + 255) / 256;        // 12500

    const float* hcur = x;
    float* hbufs[3] = { hA, hB, nullptr };
    for (int li = 0; li < 3; ++li) {
        zero4_kernel<<<zeroGrid, 256, 0, stream>>>((float4*)agg, feat4);
        scatter_kernel<<<scatterGrid, 256, 0, stream>>>(hcur, esrc, edst, agg);
        sage_gemm_kernel<<<gemmGrid, 256, 0, stream>>>(agg, cnt, hcur,
                                                       Wl[li], Wr[li], bb[li],
                                                       out + li * DIM, hbufs[li]);
        hcur = hbufs[li];
    }
}